// TokenEmbedding_80479097193165
// MI455X (gfx1250) — compile-verified
//
#include <hip/hip_runtime.h>

typedef float v2f __attribute__((ext_vector_type(2)));
typedef float v8f __attribute__((ext_vector_type(8)));

#define C_IN        7
#define D_MODEL     512
#define TAO         3
#define M_TAPS      5
#define KERNELS     73          /* D_MODEL / C_IN */
#define NFEAT       18          /* (M_TAPS+1)*3   */
#define KPAD        20          /* K padded to multiple of 4 */
#define NCOL        80          /* 73 conv + 1 leftout + pad to 5x16 */
#define NPOS        4096        /* power of two -> wrap via & (NPOS-1) */
#define NBATCH      32
#define POS_PER_BLK 128
#define THREADS     256

__global__ __launch_bounds__(THREADS)
void token_embed_wmma_f32(const float* __restrict__ x,
                          const float* __restrict__ conv_w,
                          const float* __restrict__ conv_b,
                          const float* __restrict__ leftout_w,
                          const float* __restrict__ leftout_b,
                          float* __restrict__ out)
{
    __shared__ float Bmat[KPAD * NCOL];   // [j=feature][k=output column], zero padded
    __shared__ float biasLDS[NCOL];

    const int tid = threadIdx.x;

    // ---- stage weights into LDS in GEMM-B form (shared by every channel c) ----
    for (int i = tid; i < KPAD * NCOL; i += THREADS) {
        const int j = i / NCOL;           // feature index: j = m*3 + h
        const int k = i % NCOL;           // output column
        float v = 0.f;
        if (j < NFEAT) {
            if (k < KERNELS)       v = conv_w[k * NFEAT + j];
            else if (k == KERNELS) v = leftout_w[j];
        }
        Bmat[i] = v;
    }
    if (tid < NCOL) {
        float v = 0.f;
        if (tid < KERNELS)       v = conv_b[tid];
        else if (tid == KERNELS) v = leftout_b[0];
        biasLDS[tid] = v;
    }
    __syncthreads();

    const int lane = tid & 31;
    const int wave = tid >> 5;
    const int col  = lane & 15;              // A: row M, B/D: column N
    const int khi  = (lane >> 4) << 1;       // K sub-offset held by this lane (0 or 2)

    const int b  = blockIdx.y;
    const int t0 = blockIdx.x * POS_PER_BLK + wave * 16;  // wave's 16-position tile
    const int t  = t0 + col;                              // this lane's A row position

    // ---- hoist B fragments out of the channel loop: Bfrag[nt][kc] ----
    v2f Bfrag[5][5];
    #pragma unroll
    for (int nt = 0; nt < 5; ++nt) {
        #pragma unroll
        for (int kc = 0; kc < 5; ++kc) {
            const int j0 = 4 * kc + khi;
            const int k  = nt * 16 + col;
            v2f bf;
            bf[0] = Bmat[(j0 + 0) * NCOL + k];
            bf[1] = Bmat[(j0 + 1) * NCOL + k];
            Bfrag[nt][kc] = bf;
        }
    }

    const float* xb = x + (size_t)b * NPOS * C_IN;

    for (int c = 0; c < C_IN; ++c) {
        // ---- build A fragments: branchless gather (clamped load + cndmask select) ----
        v2f Afrag[5];
        #pragma unroll
        for (int kc = 0; kc < 5; ++kc) {
            v2f af;
            #pragma unroll
            for (int v = 0; v < 2; ++v) {
                const int j = 4 * kc + khi + v;     // feature index (lane-dependent)
                const int m = j / 3;
                const int h = j % 3;
                // circular tap position, power-of-two wrap: tt = (t+h-1) mod 4096
                const int tt = (t + h - 1 + NPOS) & (NPOS - 1);
                // valid: inside feature range AND past warmup region
                const bool valid = (j < NFEAT) & (tt >= M_TAPS * TAO);
                // clamp index so the (unconditional) load is always in-bounds
                int idx = tt - m * TAO;
                idx = idx < 0 ? 0 : idx;
                const float ld = xb[(size_t)idx * C_IN + c];
                af[v] = valid ? ld : 0.f;
            }
            Afrag[kc] = af;
        }

        const int klim = (c == C_IN - 1) ? (KERNELS + 1) : KERNELS;

        #pragma unroll
        for (int nt = 0; nt < 5; ++nt) {
            const float bias = biasLDS[nt * 16 + col];
            v8f acc = { bias, bias, bias, bias, bias, bias, bias, bias };
            #pragma unroll
            for (int kc = 0; kc < 5; ++kc) {
                acc = __builtin_amdgcn_wmma_f32_16x16x4_f32(
                        /*neg_a=*/false, Afrag[kc],
                        /*neg_b=*/false, Bfrag[nt][kc],
                        /*c_mod=*/(short)0, acc,
                        /*reuse_a=*/false, /*reuse_b=*/false);
            }

            // ---- predicated nontemporal store (output is write-once streaming) ----
            const int k = nt * 16 + col;
            if (k < klim) {
                const int d = c * KERNELS + k;                 // c==6,k==73 -> d=511
                const int rowBase = t0 + ((lane >> 4) << 3);   // VGPR r holds row r (+8 hi half)
                float* op = out + (((size_t)b * NPOS) + rowBase) * D_MODEL + d;
                #pragma unroll
                for (int r = 0; r < 8; ++r)
                    __builtin_nontemporal_store(acc[r], op + (size_t)r * D_MODEL);
            }
        }
    }
}

extern "C" void kernel_launch(void* const* d_in, const int* in_sizes, int n_in,
                              void* d_out, int out_size, void* d_ws, size_t ws_size,
                              hipStream_t stream) {
    (void)in_sizes; (void)n_in; (void)d_ws; (void)ws_size; (void)out_size;
    const float* x         = (const float*)d_in[0];
    const float* conv_w    = (const float*)d_in[1];
    const float* conv_b    = (const float*)d_in[2];
    const float* leftout_w = (const float*)d_in[3];
    const float* leftout_b = (const float*)d_in[4];
    float* out = (float*)d_out;

    dim3 grid(NPOS / POS_PER_BLK, NBATCH);
    dim3 block(THREADS);
    token_embed_wmma_f32<<<grid, block, 0, stream>>>(x, conv_w, conv_b,
                                                     leftout_w, leftout_b, out);
}